// Augment_70566312673947
// MI455X (gfx1250) — compile-verified
//
#include <hip/hip_runtime.h>

// Augmentation = fused flip(W,D) . roll(5,-7) . rotate40(NN) on [B,H,W,D,F] f32.
// Pure gather, 0 FLOPs, ~256 MiB traffic -> HBM-bound (~11 us @ 23.3 TB/s).
// Strategy: one wave32 per 512B (b,h,w) block; lane = one float4.
//  - (si,sj,valid) uniform per wave -> no divergence.
//  - input (128 MiB) fits MI455X 192 MB L2 and is re-read by the NN map:
//    keep RT loads; stream output with non-temporal stores so writes don't
//    evict the L2-resident input.
//  - software-pipelined: each block's source index is computed exactly once,
//    prefetched (global_prefetch) when computed, consumed next iteration.

typedef __attribute__((ext_vector_type(4))) float v4f;

__device__ __forceinline__ int src_block_index(int blk, bool& valid)
{
    const int w = blk & 127;
    const int h = (blk >> 7) & 127;
    const int b = blk >> 14;
    const int h2 = (h + 123) & 127;       // roll shift +5 on H: (h-5) mod 128
    const int w2 = (134 - w) & 127;       // flip W then roll shift -7 on W
    const float fi = (float)h2 - 63.5f;
    const float fj = (float)w2 - 63.5f;
    const float c = 0.76604444311897801f; // cos(40 deg)
    const float s = 0.64278760968653933f; // sin(40 deg)
    const float si_f = fmaf(c, fi, fmaf(s, fj, 63.5f));
    const float sj_f = fmaf(-s, fi, fmaf(c, fj, 63.5f));
    int si = (int)rintf(si_f);            // round-half-even == jnp.round
    int sj = (int)rintf(sj_f);
    valid = ((unsigned)si < 128u) && ((unsigned)sj < 128u);
    si = min(max(si, 0), 127);
    sj = min(max(sj, 0), 127);
    return ((b << 14) | (si << 7) | sj) * 32;   // float4 index of 512B source block
}

__global__ __launch_bounds__(256)
void Augment_70566312673947_kernel(const float* __restrict__ in,
                                   float* __restrict__ out,
                                   int nblk)
{
    const v4f* __restrict__ in4 = (const v4f*)in;
    v4f* __restrict__ out4      = (v4f*)out;

    const int tid    = blockIdx.x * blockDim.x + threadIdx.x;
    const int lane   = tid & 31;
    const int wave0  = tid >> 5;
    const int nwaves = (gridDim.x * blockDim.x) >> 5;

    // lane -> source float4 slot inside the 512B block (d flipped: d2 = 7-d)
    const int d        = lane >> 2;
    const int fc       = lane & 3;
    const int src_lane = ((7 - d) << 2) | fc;

    if (wave0 >= nblk) return;

    // Prologue: compute + prefetch the first block's source.
    bool valid;
    int  sbase = src_block_index(wave0, valid);
    __builtin_prefetch((const void*)(in4 + (sbase + src_lane)), 0, 0);

    for (int blk = wave0; blk < nblk; blk += nwaves) {
        const bool cur_valid = valid;
        const int  cur_base  = sbase;

        // Compute + prefetch next iteration's source (index math done once).
        const int blk_next = blk + nwaves;
        if (blk_next < nblk) {
            sbase = src_block_index(blk_next, valid);
            __builtin_prefetch((const void*)(in4 + (sbase + src_lane)), 0, 0);
        }

        v4f v = {0.0f, 0.0f, 0.0f, 0.0f};
        if (cur_valid) {                           // wave-uniform branch
            v = in4[cur_base + src_lane];          // global_load_b128 (RT: L2-resident input)
        }
        __builtin_nontemporal_store(v, out4 + (blk * 32 + lane));  // b128 NT store
    }
}

extern "C" void kernel_launch(void* const* d_in, const int* in_sizes, int n_in,
                              void* d_out, int out_size, void* d_ws, size_t ws_size,
                              hipStream_t stream)
{
    const float* in = (const float*)d_in[0];
    float* out      = (float*)d_out;

    const int nblk = 16 * 128 * 128;   // (b,h,w) blocks of 512 B each

    // 8192 blocks x 256 threads = 65536 wave32s; 4 grid-stride iterations/wave.
    dim3 grid(8192);
    dim3 block(256);
    Augment_70566312673947_kernel<<<grid, block, 0, stream>>>(in, out, nblk);
}